// SinkhornRouter_51573967290555
// MI455X (gfx1250) — compile-verified
//
#include <hip/hip_runtime.h>
#include <math.h>
#include <stdint.h>

typedef __attribute__((ext_vector_type(2))) float v2f;
typedef __attribute__((ext_vector_type(8))) float v8f;

#define N_TOK 32768
#define D     512
#define NE    8
#define MSEL  1024
#define NBINS 2048

// ---------------- zero kernels ----------------
__global__ void zf32_kernel(float* p, int n) {
  int i = blockIdx.x * blockDim.x + threadIdx.x;
  int stride = gridDim.x * blockDim.x;
  for (; i < n; i += stride) p[i] = 0.0f;
}
__global__ void zu32_kernel(unsigned* p, int n) {
  int i = blockIdx.x * blockDim.x + threadIdx.x;
  int stride = gridDim.x * blockDim.x;
  for (; i < n; i += stride) p[i] = 0u;
}

// ---------------- gate logits: t[token*8+e] = log(max(x·gw, 1e-10)) ----------------
__global__ __launch_bounds__(128) void gate_kernel(const float* __restrict__ x,
                                                   const float* __restrict__ gw,
                                                   float* __restrict__ t) {
  __shared__ float Xs[16 * 520];
  __shared__ float Gs[D * NE];
  int tid  = threadIdx.x;
  int tok0 = blockIdx.x * 16;
  for (int i = tid; i < D * NE; i += 128) Gs[i] = gw[i];
  for (int i = tid; i < 16 * D; i += 128) {
    int r = i >> 9, c = i & 511;
    Xs[r * 520 + c] = x[(size_t)(tok0 + r) * D + c];
  }
  __syncthreads();
  int r = tid >> 3;   // 0..15
  int e = tid & 7;
  const float* xr = Xs + r * 520;
  float acc = 0.0f;
  for (int d = 0; d < D; ++d) acc = fmaf(xr[d], Gs[d * NE + e], acc);
  acc = fmaxf(acc, 1e-10f);
  t[(size_t)(tok0 + r) * NE + e] = __logf(acc);
}

// ---------------- sinkhorn: per-block partial column (max, sumexp) ----------------
__global__ __launch_bounds__(256) void col_partial_kernel(const float* __restrict__ t,
                                                          float* __restrict__ pm,
                                                          float* __restrict__ ps) {
  __shared__ float sm[256], ss[256];
  int tid  = threadIdx.x;
  int e    = tid & 7;
  int g    = tid >> 3;              // 0..31
  int base = blockIdx.x * 512;
  float m = -3.0e38f, s = 0.0f;
  for (int i = 0; i < 16; ++i) {
    float v = t[(size_t)(base + g + 32 * i) * NE + e];
    if (v > m) { s = s * __expf(m - v) + 1.0f; m = v; }
    else       { s += __expf(v - m); }
  }
  sm[tid] = m; ss[tid] = s;
  __syncthreads();
  if (tid < 8) {
    float M = sm[tid], S = ss[tid];
    for (int k = tid + 8; k < 256; k += 8) {
      float m2 = sm[k], s2 = ss[k];
      if (m2 > M) { S = S * __expf(M - m2) + s2; M = m2; }
      else        { S += s2 * __expf(m2 - M); }
    }
    pm[blockIdx.x * 8 + tid] = M;
    ps[blockIdx.x * 8 + tid] = S;
  }
}

// ------- finalize column lse (redundant per block) + fused column & row update -------
__global__ __launch_bounds__(256) void col_row_update_kernel(float* __restrict__ t,
                                                             const float* __restrict__ pm,
                                                             const float* __restrict__ ps) {
  __shared__ float clse[8];
  int tid = threadIdx.x;
  if (tid < 8) {
    float M = -3.0e38f, S = 0.0f;
    for (int b = 0; b < 64; ++b) {
      float m2 = pm[b * 8 + tid], s2 = ps[b * 8 + tid];
      if (m2 > M) { S = S * __expf(M - m2) + s2; M = m2; }
      else        { S += s2 * __expf(m2 - M); }
    }
    clse[tid] = M + __logf(S);
  }
  __syncthreads();
  int token = blockIdx.x * 256 + tid;
  float v[8];
  float rm = -3.0e38f;
  for (int e = 0; e < 8; ++e) {
    v[e] = t[(size_t)token * NE + e] - clse[e];
    rm = fmaxf(rm, v[e]);
  }
  float rs = 0.0f;
  for (int e = 0; e < 8; ++e) rs += __expf(v[e] - rm);
  float rlse = rm + __logf(rs);
  for (int e = 0; e < 8; ++e) t[(size_t)token * NE + e] = v[e] - rlse;
}

// ---------------- histogram of exp(t) per expert ----------------
__global__ __launch_bounds__(256) void hist_kernel(const float* __restrict__ t,
                                                   unsigned* __restrict__ hist) {
  int gid = blockIdx.x * 256 + threadIdx.x;   // token*8 + e
  int e = gid & 7;
  float v = __expf(t[gid]);
  int bin = (int)(v * (float)NBINS);
  bin = bin > (NBINS - 1) ? (NBINS - 1) : (bin < 0 ? 0 : bin);
  atomicAdd(&hist[e * NBINS + bin], 1u);
}

// ---------------- per-expert threshold bin: top-1024 ----------------
__global__ void thresh_kernel(const unsigned* __restrict__ hist,
                              int* __restrict__ tbin, int* __restrict__ needt) {
  int tid = threadIdx.x;
  if (tid < NE) {
    unsigned cum = 0;
    int tb = 0, need = MSEL;
    for (int b = NBINS - 1; b >= 0; --b) {
      unsigned c = hist[tid * NBINS + b];
      if (cum + c >= (unsigned)MSEL) { tb = b; need = MSEL - (int)cum; break; }
      cum += c;
    }
    tbin[tid] = tb;
    needt[tid] = need;
  }
}

// -------- select elements strictly above threshold bin; compact routed list --------
__global__ __launch_bounds__(256) void select_main_kernel(const float* __restrict__ t,
                                                          const int* __restrict__ tbin,
                                                          unsigned* __restrict__ selc,
                                                          unsigned* __restrict__ routed,
                                                          unsigned* __restrict__ rcnt) {
  int gid = blockIdx.x * 256 + threadIdx.x;
  int e = gid & 7;
  int token = gid >> 3;
  float v = __expf(t[gid]);
  int bin = (int)(v * (float)NBINS);
  bin = bin > (NBINS - 1) ? (NBINS - 1) : (bin < 0 ? 0 : bin);
  if (bin > tbin[e]) {
    atomicAdd(&selc[token], 1u);
    if (v > 0.5f) {
      unsigned p = atomicAdd(&rcnt[e], 1u);
      routed[e * MSEL + p] = (unsigned)token;
    }
  }
}

// -------- deterministic tie selection in threshold bin (by token index) --------
__global__ __launch_bounds__(256) void select_ties_kernel(const float* __restrict__ t,
                                                          const int* __restrict__ tbin,
                                                          const int* __restrict__ needt,
                                                          unsigned* __restrict__ selc,
                                                          unsigned* __restrict__ routed,
                                                          unsigned* __restrict__ rcnt) {
  int tid  = threadIdx.x;
  int wave = tid >> 5;      // one wave per expert
  int lane = tid & 31;
  if (wave >= NE) return;
  int e    = wave;
  int tb   = tbin[e];
  int need = needt[e];
  int taken = 0;
  for (int base = 0; base < N_TOK; base += 32) {
    int token = base + lane;
    float v = __expf(t[(size_t)token * NE + e]);
    int bin = (int)(v * (float)NBINS);
    bin = bin > (NBINS - 1) ? (NBINS - 1) : (bin < 0 ? 0 : bin);
    bool isTie = (bin == tb);
    unsigned mask = (unsigned)__ballot(isTie);
    int pre = __popc(mask & ((1u << lane) - 1u));
    if (isTie && (taken + pre < need)) {
      atomicAdd(&selc[token], 1u);
      if (v > 0.5f) {
        unsigned p = atomicAdd(&rcnt[e], 1u);
        routed[e * MSEL + p] = (unsigned)token;
      }
    }
    taken += __popc(mask);
    if (taken >= need) break;
  }
}

// ---------------- expert GEMM via V_WMMA_F32_16X16X4_F32 ----------------
// grid = 8 experts * 64 m-tiles; block = 256 (8 waves).
// A tile (16 gathered token rows) staged via GLOBAL_LOAD_ASYNC_TO_LDS_B128.
// Each wave keeps TWO independent accumulators (tiles nt, nt+8): breaks the
// WMMA C-operand dependency chain and reuses each A fragment for 2 WMMAs.
__global__ __launch_bounds__(256) void expert_gemm_wmma_kernel(
    const float* __restrict__ x, const float* __restrict__ experts,
    const unsigned* __restrict__ routed, const unsigned* __restrict__ rcnt,
    float* __restrict__ out) {
  __shared__ float Ald[16 * 516];   // row stride 516 -> conflict-free column reads
  __shared__ int   stok[16];
  int e     = blockIdx.x >> 6;
  int mtile = blockIdx.x & 63;
  int cnt   = (int)rcnt[e];
  int ntile = (cnt + 15) >> 4;
  if (mtile >= ntile) return;
  int tid = threadIdx.x;
  if (tid < 16) {
    int gi = mtile * 16 + tid;
    stok[tid] = (gi < cnt) ? (int)routed[e * MSEL + gi] : -1;
  }
  __syncthreads();

  // Stage A: 16 rows x 512 f32 in 16B chunks. Valid rows go memory->LDS via the
  // CDNA5 async DMA path (ASYNCcnt); pad rows are zero-filled with DS stores.
  for (int i = tid; i < 16 * 128; i += 256) {
    int r  = i >> 7;            // row 0..15
    int c4 = (i & 127) << 2;    // starting float within row (16B aligned)
    int tok = stok[r];
    float* ldst = &Ald[r * 516 + c4];
    if (tok >= 0) {
      unsigned lds_off = (unsigned)(uintptr_t)ldst;  // low 32 bits = LDS byte offset
      const float* g = x + (size_t)tok * D + c4;
      asm volatile("global_load_async_to_lds_b128 %0, %1, off"
                   :: "v"(lds_off), "v"(g) : "memory");
    } else {
      float4 z = make_float4(0.f, 0.f, 0.f, 0.f);
      *(float4*)ldst = z;
    }
  }
  asm volatile("s_wait_asynccnt 0x0" ::: "memory");
  __syncthreads();

  const float* W = experts + (size_t)e * D * D;
  int lane   = tid & 31;
  int wave   = tid >> 5;
  int half   = lane >> 4;     // 0: K+0..1 / rows 0..7 ; 1: K+2..3 / rows 8..15
  int colOff = lane & 15;
  int rowA   = lane & 15;
  const float* Arow = Ald + rowA * 516 + 2 * half;

  for (int pass = 0; pass < 2; ++pass) {
    int nt = wave + pass * 16;          // this wave's tile pair: nt and nt+8
    int n0 = nt << 4;
    int n1 = n0 + 128;                  // (nt+8) << 4
    const float* Wn0 = W + n0 + colOff;
    const float* Wn1 = W + n1 + colOff;
    v8f acc0 = {};
    v8f acc1 = {};
#pragma unroll 4
    for (int k0 = 0; k0 < D; k0 += 4) {
      v2f a, b0, b1;
      // A 16x4 frag: lanes 0-15 rows M=0..15, vgpr j holds K=k0+2*half+j
      a.x = Arow[k0];
      a.y = Arow[k0 + 1];
      // B 4x16 frags: vgpr j holds K row k0+2*half+j, lanes give N column
      const float* wk0 = Wn0 + (size_t)(k0 + 2 * half) * D;
      const float* wk1 = Wn1 + (size_t)(k0 + 2 * half) * D;
      b0.x = wk0[0]; b0.y = wk0[D];
      b1.x = wk1[0]; b1.y = wk1[D];
      acc0 = __builtin_amdgcn_wmma_f32_16x16x4_f32(false, a, false, b0,
                                                   (short)0, acc0, false, false);
      acc1 = __builtin_amdgcn_wmma_f32_16x16x4_f32(false, a, false, b1,
                                                   (short)0, acc1, false, false);
    }
    // C/D layout: vgpr v holds row v + 8*half, lanes give column
    for (int v = 0; v < 8; ++v) {
      int row = v + 8 * half;
      int tok = stok[row];
      if (tok >= 0) {
        atomicAdd(&out[(size_t)tok * D + n0 + colOff], acc0[v]);
        atomicAdd(&out[(size_t)tok * D + n1 + colOff], acc1[v]);
      }
    }
  }
}

// ---------------- final divide by (sel_count + 1) ----------------
__global__ void finalize_kernel(float* __restrict__ out,
                                const unsigned* __restrict__ selc) {
  int i = blockIdx.x * blockDim.x + threadIdx.x;
  int stride = gridDim.x * blockDim.x;
  for (; i < N_TOK * D; i += stride) {
    int token = i >> 9;
    out[i] = out[i] / (float)(selc[token] + 1u);
  }
}

static inline size_t align256(size_t v) { return (v + 255) & ~(size_t)255; }

extern "C" void kernel_launch(void* const* d_in, const int* in_sizes, int n_in,
                              void* d_out, int out_size, void* d_ws, size_t ws_size,
                              hipStream_t stream) {
  (void)in_sizes; (void)n_in; (void)out_size; (void)ws_size;
  const float* x       = (const float*)d_in[0];   // (4, 8192, 512)
  const float* gate_w  = (const float*)d_in[1];   // (1, 512, 8)
  const float* experts = (const float*)d_in[2];   // (8, 512, 512)
  float* out = (float*)d_out;                     // (4, 8192, 512) == (N_TOK, D)

  char* w = (char*)d_ws;
  size_t off = 0;
  float*    t      = (float*)(w + off); off = align256(off + (size_t)N_TOK * NE * 4);
  float*    pm     = (float*)(w + off); off = align256(off + 64 * NE * 4);
  float*    ps     = (float*)(w + off); off = align256(off + 64 * NE * 4);
  unsigned* hist   = (unsigned*)(w + off); off = align256(off + (size_t)NE * NBINS * 4);
  unsigned* selc   = (unsigned*)(w + off); off = align256(off + (size_t)N_TOK * 4);
  unsigned* routed = (unsigned*)(w + off); off = align256(off + (size_t)NE * MSEL * 4);
  unsigned* rcnt   = (unsigned*)(w + off); off = align256(off + NE * 4);
  int*      tbin   = (int*)(w + off); off = align256(off + NE * 4);
  int*      needt  = (int*)(w + off); off = align256(off + NE * 4);

  // per-call scratch init (harness does not re-zero between replays)
  zf32_kernel<<<4096, 256, 0, stream>>>(out, N_TOK * D);
  zu32_kernel<<<64, 256, 0, stream>>>(hist, NE * NBINS);
  zu32_kernel<<<64, 256, 0, stream>>>(selc, N_TOK);
  zu32_kernel<<<1, 32, 0, stream>>>(rcnt, NE);

  // gate logits -> log
  gate_kernel<<<N_TOK / 16, 128, 0, stream>>>(x, gate_w, t);

  // sinkhorn, 8 iterations (column lse then fused column+row update)
  for (int it = 0; it < 8; ++it) {
    col_partial_kernel<<<64, 256, 0, stream>>>(t, pm, ps);
    col_row_update_kernel<<<128, 256, 0, stream>>>(t, pm, ps);
  }

  // top-1024 per expert via histogram select
  hist_kernel<<<(N_TOK * NE) / 256, 256, 0, stream>>>(t, hist);
  thresh_kernel<<<1, 32, 0, stream>>>(hist, tbin, needt);
  select_main_kernel<<<(N_TOK * NE) / 256, 256, 0, stream>>>(t, tbin, selc, routed, rcnt);
  select_ties_kernel<<<1, 256, 0, stream>>>(t, tbin, needt, selc, routed, rcnt);

  // expert GEMMs (f32 WMMA, async-LDS A staging) with atomic scatter into out
  expert_gemm_wmma_kernel<<<NE * 64, 256, 0, stream>>>(x, experts, routed, rcnt, out);

  // divide by (selection count + 1)
  finalize_kernel<<<8192, 256, 0, stream>>>(out, selc);
}